// DeeperGCN_65060164600379
// MI455X (gfx1250) — compile-verified
//
#include <hip/hip_runtime.h>
#include <math.h>

typedef float v2f __attribute__((ext_vector_type(2)));
typedef float v8f __attribute__((ext_vector_type(8)));

#define LN_EPS  1e-5f
#define EPS_MSG 1e-7f
#define EPS_SM  1e-16f

__device__ __forceinline__ float wave16_sum(float v) {
#pragma unroll
  for (int m = 1; m < 16; m <<= 1) v += __shfl_xor(v, m, 32);
  return v;
}
__device__ __forceinline__ float wave32_sum(float v) {
#pragma unroll
  for (int m = 1; m < 32; m <<= 1) v += __shfl_xor(v, m, 32);
  return v;
}
// fp32 atomic max via integer ordering trick (global_atomic_max_i32 / umin)
__device__ __forceinline__ void atomicMaxF(float* a, float v) {
  if (v >= 0.0f) atomicMax((int*)a, __float_as_int(v));
  else           atomicMin((unsigned int*)a, __float_as_uint(v));
}
__device__ __forceinline__ v8f splat8(float x) {
  v8f v;
#pragma unroll
  for (int i = 0; i < 8; ++i) v[i] = x;
  return v;
}

// ---------------------------------------------------------------------------
// Node prep: h = (do_ln ? relu(LN(x)) : x); msg = relu(h)+eps; init mx/sm/agg.
// One wave (32 lanes) per node; each lane owns channels c and c+32.
// ---------------------------------------------------------------------------
__global__ __launch_bounds__(256) void prep_kernel(
    const float* __restrict__ xin, const float* __restrict__ g,
    const float* __restrict__ b, int do_ln,
    float* __restrict__ h, float* __restrict__ msgn,
    float* __restrict__ mx, float* __restrict__ sm, float* __restrict__ agg,
    int n)
{
  int node = blockIdx.x * 8 + (threadIdx.x >> 5);
  if (node >= n) return;
  int lane = threadIdx.x & 31;
  int i0 = node * 64 + lane, i1 = i0 + 32;
  float v0 = xin[i0], v1 = xin[i1];
  float h0, h1;
  if (do_ln) {
    float s  = wave32_sum(v0 + v1);
    float q  = wave32_sum(v0 * v0 + v1 * v1);
    float mu = s * (1.0f / 64.0f);
    float rs = rsqrtf(q * (1.0f / 64.0f) - mu * mu + LN_EPS);
    h0 = fmaxf((v0 - mu) * rs * g[lane]      + b[lane],      0.0f);
    h1 = fmaxf((v1 - mu) * rs * g[lane + 32] + b[lane + 32], 0.0f);
  } else { h0 = v0; h1 = v1; }
  h[i0] = h0;  h[i1] = h1;
  msgn[i0] = fmaxf(h0, 0.0f) + EPS_MSG;
  msgn[i1] = fmaxf(h1, 0.0f) + EPS_MSG;
  float ninf = -__builtin_inff();
  mx[i0] = ninf;  mx[i1] = ninf;
  sm[i0] = 0.0f;  sm[i1] = 0.0f;
  agg[i0] = 0.0f; agg[i1] = 0.0f;
}

// ---------------------------------------------------------------------------
// Edge passes: one thread per (edge, channel). All arrays are L2-resident.
// ---------------------------------------------------------------------------
__global__ __launch_bounds__(256) void edge_max_kernel(
    const float* __restrict__ msgn, const int* __restrict__ src,
    const int* __restrict__ dst, const float* __restrict__ t, int layer,
    float* __restrict__ mx, int E)
{
  int gid = blockIdx.x * 256 + threadIdx.x;
  int e = gid >> 6;
  if (e >= E) return;
  int c = gid & 63;
  int s = src[e], d = dst[e];
  float w = msgn[s * 64 + c] * t[layer];
  atomicMaxF(&mx[d * 64 + c], w);
}

__global__ __launch_bounds__(256) void edge_sum_kernel(
    const float* __restrict__ msgn, const int* __restrict__ src,
    const int* __restrict__ dst, const float* __restrict__ t, int layer,
    const float* __restrict__ mx, float* __restrict__ sm, int E)
{
  int gid = blockIdx.x * 256 + threadIdx.x;
  int e = gid >> 6;
  if (e >= E) return;
  int c = gid & 63;
  int s = src[e], d = dst[e];
  int di = d * 64 + c;
  float w = msgn[s * 64 + c] * t[layer];
  atomicAdd(&sm[di], expf(w - mx[di]));
}

__global__ __launch_bounds__(256) void edge_agg_kernel(
    const float* __restrict__ msgn, const int* __restrict__ src,
    const int* __restrict__ dst, const float* __restrict__ t, int layer,
    const float* __restrict__ mx, const float* __restrict__ sm,
    float* __restrict__ agg, int E)
{
  int gid = blockIdx.x * 256 + threadIdx.x;
  int e = gid >> 6;
  if (e >= E) return;
  int c = gid & 63;
  int s = src[e], d = dst[e];
  int di = d * 64 + c;
  float mv = msgn[s * 64 + c];
  float w  = mv * t[layer];
  float al = expf(w - mx[di]) / (sm[di] + EPS_SM);
  atomicAdd(&agg[di], mv * al);
}

// ---------------------------------------------------------------------------
// MLP per 16-node tile, one wave per tile, 8 waves per block.
// out = agg + h; h1 = relu(LN(out@W1^T + b1)); y = h1@W2^T + b2 (+ residual)
// fp32 WMMA 16x16x4.  LDS: [0,8192) W1 B-frags, [8192,16384) W2 B-frags,
// [16384, +8*2112) per-wave 16x(stride 132) tile buffer.
// ---------------------------------------------------------------------------
__global__ __launch_bounds__(256) void mlp_kernel(
    const float* __restrict__ hbuf, const float* __restrict__ agg,
    const float* __restrict__ W1, const float* __restrict__ b1,
    const float* __restrict__ mg, const float* __restrict__ mb,
    const float* __restrict__ W2, const float* __restrict__ b2,
    float* __restrict__ xres, int residual, int ntiles)
{
  extern __shared__ float lds[];
  const int tid  = threadIdx.x;
  const int lane = tid & 31;
  const int wave = tid >> 5;

  // Swizzle W1 (h1[m,n] = sum_k out[m,k]*W1[n,k]) into B-fragment order:
  // frag(kt<16, ct<8): lds[((kt*8+ct)*32 + lane)*2 + r] = W1[n*64 + k]
  //   n = ct*16 + (lane&15); k = kt*4 + r + 2*(lane>=16)
  for (int i = tid; i < 8192; i += 256) {
    int r = i & 1, ln = (i >> 1) & 31, f = i >> 6;
    int ct = f & 7, kt = f >> 3;
    int nn = ct * 16 + (ln & 15);
    int kk = kt * 4 + r + ((ln >> 4) << 1);
    lds[i] = W1[nn * 64 + kk];
  }
  // W2 frags: frag(kt<32, ct<4) at +8192;  y[m,n] = sum_k h1[m,k]*W2[n,k]
  for (int i = tid; i < 8192; i += 256) {
    int r = i & 1, ln = (i >> 1) & 31, f = i >> 6;
    int ct = f & 3, kt = f >> 2;
    int nn = ct * 16 + (ln & 15);
    int kk = kt * 4 + r + ((ln >> 4) << 1);
    lds[8192 + i] = W2[nn * 128 + kk];
  }
  __syncthreads();

  int tile = blockIdx.x * 8 + wave;
  if (tile >= ntiles) return;

  float* tbuf = lds + 16384 + wave * 2112;  // 16 rows x stride 132 dwords
  const int base = tile * 16 * 64;

  // Stage out = agg + h (coalesced global, conflict-free LDS)
  for (int i = lane; i < 1024; i += 32) {
    int m = i >> 6, c = i & 63;
    tbuf[m * 132 + c] = agg[base + i] + hbuf[base + i];
  }

  const int lm   = lane & 15;
  const int koff = (lane >> 4) << 1;        // K sub-offset: 0 or 2
  const int mhi  = (lane >> 4) << 3;        // C-layout row offset: 0 or 8

  // ---- GEMM1: 16x64 @ 64x128 ----
  v2f afr[16];
#pragma unroll
  for (int kt = 0; kt < 16; ++kt)
    afr[kt] = *(const v2f*)&tbuf[lm * 132 + kt * 4 + koff];

  v8f acc[8];
#pragma unroll
  for (int ct = 0; ct < 8; ++ct) {
    v8f a = splat8(b1[ct * 16 + lm]);
#pragma unroll
    for (int kt = 0; kt < 16; ++kt) {
      v2f bfr = *(const v2f*)&lds[((kt * 8 + ct) * 32 + lane) * 2];
      a = __builtin_amdgcn_wmma_f32_16x16x4_f32(false, afr[kt], false, bfr,
                                                (short)0, a, false, false);
    }
    acc[ct] = a;
  }

  // ---- LayerNorm over 128 per row (per-r partials + 16-lane reductions) ----
  float mean[8], rstd[8];
  {
    float s[8], q[8];
#pragma unroll
    for (int r = 0; r < 8; ++r) { s[r] = 0.0f; q[r] = 0.0f; }
#pragma unroll
    for (int ct = 0; ct < 8; ++ct)
#pragma unroll
      for (int r = 0; r < 8; ++r) { float v = acc[ct][r]; s[r] += v; q[r] += v * v; }
#pragma unroll
    for (int r = 0; r < 8; ++r) {
      float ss = wave16_sum(s[r]);
      float qq = wave16_sum(q[r]);
      float mu = ss * (1.0f / 128.0f);
      mean[r] = mu;
      rstd[r] = rsqrtf(qq * (1.0f / 128.0f) - mu * mu + LN_EPS);
    }
  }
  // normalize + relu, spill h1 tile to LDS in row-major (stride 132)
#pragma unroll
  for (int ct = 0; ct < 8; ++ct) {
    float gg = mg[ct * 16 + lm], bb = mb[ct * 16 + lm];
#pragma unroll
    for (int r = 0; r < 8; ++r) {
      float v = (acc[ct][r] - mean[r]) * rstd[r] * gg + bb;
      tbuf[(r + mhi) * 132 + ct * 16 + lm] = fmaxf(v, 0.0f);
    }
  }

  // ---- GEMM2: 16x128 @ 128x64 ----
  v8f acc2[4];
#pragma unroll
  for (int ct = 0; ct < 4; ++ct) acc2[ct] = splat8(b2[ct * 16 + lm]);
#pragma unroll
  for (int kt = 0; kt < 32; ++kt) {
    v2f a2 = *(const v2f*)&tbuf[lm * 132 + kt * 4 + koff];
#pragma unroll
    for (int ct = 0; ct < 4; ++ct) {
      v2f bfr = *(const v2f*)&lds[8192 + ((kt * 4 + ct) * 32 + lane) * 2];
      acc2[ct] = __builtin_amdgcn_wmma_f32_16x16x4_f32(false, a2, false, bfr,
                                                       (short)0, acc2[ct], false, false);
    }
  }

  // ---- writeback (+ residual) ----
#pragma unroll
  for (int ct = 0; ct < 4; ++ct)
#pragma unroll
    for (int r = 0; r < 8; ++r) {
      int idx = base + (r + mhi) * 64 + ct * 16 + lm;
      float v = acc2[ct][r];
      if (residual) v += xres[idx];
      xres[idx] = v;
    }
}

// ---------------------------------------------------------------------------
// Final: out = relu(LN(x; lng[0], lnb[0]))
// ---------------------------------------------------------------------------
__global__ __launch_bounds__(256) void final_kernel(
    const float* __restrict__ xin, const float* __restrict__ g,
    const float* __restrict__ b, float* __restrict__ out, int n)
{
  int node = blockIdx.x * 8 + (threadIdx.x >> 5);
  if (node >= n) return;
  int lane = threadIdx.x & 31;
  int i0 = node * 64 + lane, i1 = i0 + 32;
  float v0 = xin[i0], v1 = xin[i1];
  float s  = wave32_sum(v0 + v1);
  float q  = wave32_sum(v0 * v0 + v1 * v1);
  float mu = s * (1.0f / 64.0f);
  float rs = rsqrtf(q * (1.0f / 64.0f) - mu * mu + LN_EPS);
  out[i0] = fmaxf((v0 - mu) * rs * g[lane]      + b[lane],      0.0f);
  out[i1] = fmaxf((v1 - mu) * rs * g[lane + 32] + b[lane + 32], 0.0f);
}

extern "C" void kernel_launch(void* const* d_in, const int* in_sizes, int n_in,
                              void* d_out, int out_size, void* d_ws, size_t ws_size,
                              hipStream_t stream) {
  (void)n_in; (void)out_size; (void)ws_size;
  const float* x   = (const float*)d_in[0];
  const int*   ei  = (const int*)  d_in[1];
  const float* t   = (const float*)d_in[2];
  const float* W1  = (const float*)d_in[3];
  const float* b1  = (const float*)d_in[4];
  const float* mg  = (const float*)d_in[5];
  const float* mb  = (const float*)d_in[6];
  const float* W2  = (const float*)d_in[7];
  const float* b2  = (const float*)d_in[8];
  const float* lng = (const float*)d_in[9];
  const float* lnb = (const float*)d_in[10];

  const int N  = in_sizes[0] / 64;
  const int E  = in_sizes[1] / 2;
  const int L  = in_sizes[2];
  const int NH = N * 64;

  const int* src = ei;
  const int* dst = ei + E;

  float* h    = (float*)d_ws;      // [N,64] conv input (post pre-norm)
  float* msgn = h    + NH;         // relu(h)+eps  (per SOURCE node)
  float* mx   = msgn + NH;         // segment max
  float* sm   = mx   + NH;         // segment sum of exp
  float* agg  = sm   + NH;         // weighted aggregate
  float* xres = agg  + NH;         // running residual state

  dim3 blk(256);
  int nodeGrid = (N + 7) / 8;
  int edgeGrid = (E * 64 + 255) / 256;
  int ntiles   = (N + 15) / 16;    // N assumed multiple of 16 (50000 = 3125*16)
  int mlpGrid  = (ntiles + 7) / 8;
  size_t ldsBytes = (16384 + 8 * 2112) * sizeof(float);  // 133120 B

  for (int i = 0; i < L; ++i) {
    const float* in = (i == 0) ? x : xres;
    prep_kernel<<<nodeGrid, blk, 0, stream>>>(in, lng + i * 64, lnb + i * 64,
                                              (i > 0) ? 1 : 0,
                                              h, msgn, mx, sm, agg, N);
    edge_max_kernel<<<edgeGrid, blk, 0, stream>>>(msgn, src, dst, t, i, mx, E);
    edge_sum_kernel<<<edgeGrid, blk, 0, stream>>>(msgn, src, dst, t, i, mx, sm, E);
    edge_agg_kernel<<<edgeGrid, blk, 0, stream>>>(msgn, src, dst, t, i, mx, sm, agg, E);
    mlp_kernel<<<mlpGrid, blk, ldsBytes, stream>>>(h, agg,
        W1 + i * 128 * 64, b1 + i * 128, mg + i * 128, mb + i * 128,
        W2 + i * 64 * 128, b2 + i * 64, xres, (i > 0) ? 1 : 0, ntiles);
  }
  final_kernel<<<nodeGrid, blk, 0, stream>>>(xres, lng, lnb, (float*)d_out, N);
}